// ASA_20727512171058
// MI455X (gfx1250) — compile-verified
//
#include <hip/hip_runtime.h>
#include <hip/hip_bf16.h>

// ---------------------------------------------------------------------------
// ASA (axial self-attention) for MI455X / gfx1250.
// f32 WMMA 16x16x4 for all matmuls, async-to-LDS DMA staging of K/V operands
// with pipelined (partial) ASYNCcnt waits, deterministic online-softmax stats.
// B=4, C=64, F=256, T=256, D_C=16, cache T=64, concat T = 320.
// ---------------------------------------------------------------------------

typedef __attribute__((ext_vector_type(2))) float v2f;
typedef __attribute__((ext_vector_type(8))) float v8f;

static __device__ __forceinline__ v8f wmma4(v2f a, v2f b, v8f c) {
  // D = A(16x4) * B(4x16) + C(16x16), f32
  return __builtin_amdgcn_wmma_f32_16x16x4_f32(false, a, false, b,
                                               (short)0, c, false, false);
}

// Async 16B copy global -> LDS (ASYNCcnt-tracked DMA path, overlaps compute).
static __device__ __forceinline__ void async_copy_b128(void* lds_ptr,
                                                       const void* gptr) {
  unsigned lds = (unsigned)(size_t)lds_ptr;            // low 32 bits = LDS addr
  unsigned long long g = (unsigned long long)(size_t)gptr;
  asm volatile("global_load_async_to_lds_b128 %0, %1, off"
               :: "v"(lds), "v"(g) : "memory");
}
static __device__ __forceinline__ void async_wait0() {
  asm volatile("s_wait_asynccnt 0" ::: "memory");
}
static __device__ __forceinline__ void async_wait16() {
  // async loads retire in order: <=16 outstanding => first 16 ops complete
  asm volatile("s_wait_asynccnt 16" ::: "memory");
}

#define B_  4
#define C_  64
#define F_  256
#define T_  256
#define DC_ 16
#define TY_ 320   // 64 cache + 256
#define EPS_ 1e-5f

// ---------------------------------------------------------------------------
// K0: fold BN params: scale = g*rsqrt(v+eps), shift = b - g*m*rsqrt(v+eps)
// channels 0..47 = fqkv, 48..79 = tqk ; proj 0..63 separately
// ---------------------------------------------------------------------------
__global__ void k_params(const float* g48, const float* b48, const float* m48,
                         const float* v48, const float* a48,
                         const float* g32, const float* b32, const float* m32,
                         const float* v32, const float* a32,
                         const float* gp, const float* bp, const float* mp,
                         const float* vp, const float* ap,
                         float* scaleA, float* shiftA, float* alphaA,
                         float* scaleP, float* shiftP, float* alphaP) {
  int i = threadIdx.x;
  if (i < 80) {
    float g, bb, mm, vv, aa;
    if (i < 48) { g = g48[i]; bb = b48[i]; mm = m48[i]; vv = v48[i]; aa = a48[i]; }
    else { int j = i - 48; g = g32[j]; bb = b32[j]; mm = m32[j]; vv = v32[j]; aa = a32[j]; }
    float r = rsqrtf(vv + EPS_);
    scaleA[i] = g * r;
    shiftA[i] = bb - g * mm * r;
    alphaA[i] = aa;
  }
  if (i < 64) {
    float r = rsqrtf(vp[i] + EPS_);
    scaleP[i] = gp[i] * r;
    shiftP[i] = bp[i] - gp[i] * mp[i] * r;
    alphaP[i] = ap[i];
  }
}

// ---------------------------------------------------------------------------
// K1: fused QKV/QK conv: GEMM M=80(ch) x K=64(C) x N=B*F*T, + BN + PReLU.
// Each wave: one 16(M)x16(N) tile, 16 WMMA steps over K.
// Channel tiles: 0 qf, 1 kf, 2 v  -> stored TRANSPOSED (b,t,c,f) so the
//   f-attention reads/DMAs are contiguous;  3 qt (b,c,f,t);
//   4 kt -> ktf (b,c,f,320) at T offset 64.
// ---------------------------------------------------------------------------
__global__ void k_qkv(const float* __restrict__ inp,
                      const float* __restrict__ wf,   // (48,64)
                      const float* __restrict__ wt,   // (32,64)
                      const float* __restrict__ scaleA,
                      const float* __restrict__ shiftA,
                      const float* __restrict__ alphaA,
                      float* __restrict__ qfT, float* __restrict__ kfT,
                      float* __restrict__ vvT, float* __restrict__ qt,
                      float* __restrict__ ktf) {
  const int wid = threadIdx.x >> 5, lane = threadIdx.x & 31;
  const int li = lane & 15, half = lane >> 4;
  int gid = blockIdx.x * 8 + wid;          // 81920 wave tiles
  int mt = gid % 5;
  int nt = gid / 5;
  int n0 = nt * 16;
  int t0 = n0 & 255, f = (n0 >> 8) & 255, b = n0 >> 16;

  const float* W; int chBase;
  if (mt < 3) { W = wf; chBase = mt * 16; } else { W = wt; chBase = (mt - 3) * 16; }

  v8f acc = {};
  for (int kw = 0; kw < 16; ++kw) {
    int c0 = 4 * kw + 2 * half;
    v2f a, bv;
    a.x = W[(chBase + li) * 64 + c0];
    a.y = W[(chBase + li) * 64 + c0 + 1];
    const float* ip = inp + (size_t)b * C_ * F_ * T_ + (size_t)f * T_ + t0 + li;
    bv.x = ip[(size_t)c0 * (F_ * T_)];
    bv.y = ip[(size_t)(c0 + 1) * (F_ * T_)];
    acc = wmma4(a, bv, acc);
  }

  int ch0 = mt * 16;
  for (int r = 0; r < 8; ++r) {
    int row = ch0 + r + 8 * half;   // global channel 0..79
    int cc  = row - ch0;            // local 0..15
    float y = acc[r] * scaleA[row] + shiftA[row];
    y = (y >= 0.f) ? y : alphaA[row] * y;
    size_t t = (size_t)t0 + li;
    if (mt == 4) {
      ktf[(((size_t)b * DC_ + cc) * F_ + f) * TY_ + 64 + t] = y;
    } else if (mt == 3) {
      qt[(((size_t)b * DC_ + cc) * F_ + f) * T_ + t] = y;
    } else {
      float* d = (mt == 0) ? qfT : (mt == 1) ? kfT : vvT;
      d[(((size_t)b * T_ + t) * DC_ + cc) * F_ + f] = y;     // (b,t,c,f)
    }
  }
}

// ---------------------------------------------------------------------------
// K2: copy cache -> first 64 T slots. cache (B,32,F,64): ch<16 k, ch>=16 v.
// ---------------------------------------------------------------------------
__global__ void k_cachein(const float* __restrict__ cache,
                          float* __restrict__ ktf, float* __restrict__ fof) {
  size_t i = (size_t)blockIdx.x * 256 + threadIdx.x;
  const size_t N = (size_t)B_ * 32 * F_ * 64;
  if (i >= N) return;
  int tc = (int)(i & 63); size_t r = i >> 6;
  int f = (int)(r & 255); r >>= 8;
  int ch = (int)(r & 31); int b = (int)(r >> 5);
  float val = cache[i];
  float* dst = (ch < 16) ? ktf : fof;
  int cc = ch & 15;
  dst[(((size_t)b * DC_ + cc) * F_ + f) * TY_ + tc] = val;
}

// ---------------------------------------------------------------------------
// K3: frequency attention. Block = 64 threads (2 waves); block = (b, t, 32 f).
// Async-DMA the K tile then the V tile into LDS; partial wait (<=16) releases
// the K tile for the score WMMAs while the V DMA still runs under softmax.
// Wave owns 16 f rows: scores (16x256) -> LDS, softmax, then out = V * P^T.
// ---------------------------------------------------------------------------
__global__ void k_fattn(const float* __restrict__ qfT,
                        const float* __restrict__ kfT,
                        const float* __restrict__ vvT,
                        float* __restrict__ fof) {
  __shared__ float P[2][16][256];   // 32 KB
  __shared__ float Ks[16][256];     // 16 KB
  __shared__ float Vs[16][256];     // 16 KB   (total 64 KB << 320 KB WGP LDS)
  const int wid = threadIdx.x >> 5, lane = threadIdx.x & 31;
  const int li = lane & 15, half = lane >> 4;
  int fb = blockIdx.x & 7;
  int t  = (blockIdx.x >> 3) & 255;
  int b  = blockIdx.x >> 11;
  int f0 = fb * 32 + wid * 16;

  // kick off async DMA: K tile first (needed first), then V tile.
  // 16x256 floats each, rows contiguous; 16 b128 wave-ops per tile per wave.
  {
    const float* ksrc = kfT + ((size_t)b * T_ + t) * DC_ * F_;
    const float* vsrc = vvT + ((size_t)b * T_ + t) * DC_ * F_;
    for (int i = 0; i < 16; ++i) {
      int e4 = threadIdx.x + 64 * i;
      int c = e4 >> 6, y4 = e4 & 63;
      async_copy_b128(&Ks[c][y4 * 4], ksrc + (size_t)c * F_ + y4 * 4);
    }
    for (int i = 0; i < 16; ++i) {
      int e4 = threadIdx.x + 64 * i;
      int c = e4 >> 6, y4 = e4 & 63;
      async_copy_b128(&Vs[c][y4 * 4], vsrc + (size_t)c * F_ + y4 * 4);
    }
  }

  // Q operand (coalesced in f thanks to (b,t,c,f) layout), overlaps the DMA
  v2f qa[4];
  for (int kw = 0; kw < 4; ++kw) {
    int c0 = 4 * kw + 2 * half;
    qa[kw].x = qfT[(((size_t)b * T_ + t) * DC_ + c0)     * F_ + f0 + li];
    qa[kw].y = qfT[(((size_t)b * T_ + t) * DC_ + c0 + 1) * F_ + f0 + li];
  }
  async_wait16();       // K tile resident (in-order completion)
  __syncthreads();

  // scores: S[f,y] = 0.25 * sum_c Q[c,f] K[c,y] ; B operand from LDS
  for (int yt = 0; yt < 16; ++yt) {
    v8f acc = {};
    for (int kw = 0; kw < 4; ++kw) {
      int c0 = 4 * kw + 2 * half;
      v2f bv;
      bv.x = Ks[c0][yt * 16 + li];
      bv.y = Ks[c0 + 1][yt * 16 + li];
      acc = wmma4(qa[kw], bv, acc);
    }
    for (int r = 0; r < 8; ++r)
      P[wid][r + 8 * half][yt * 16 + li] = acc[r] * 0.25f;
  }
  __syncthreads();

  // softmax over y per row: lane L handles row L&15, column half L>>4
  {
    float* row = &P[wid][li][half * 128];
    float m = -3.4e38f;
    for (int j = 0; j < 128; ++j) m = fmaxf(m, row[j]);
    m = fmaxf(m, __shfl_xor(m, 16));
    float s = 0.f;
    for (int j = 0; j < 128; ++j) { float e = __expf(row[j] - m); row[j] = e; s += e; }
    s += __shfl_xor(s, 16);
    float inv = 1.f / s;
    for (int j = 0; j < 128; ++j) row[j] *= inv;
  }
  async_wait0();        // V tile resident before anyone reads Vs
  __syncthreads();

  // out[c,f] = sum_y V[c,y] P[f,y]  (M=c, N=f, K=y=256 -> 64 WMMA, A from LDS)
  v8f acc = {};
  for (int kw = 0; kw < 64; ++kw) {
    int y0 = 4 * kw + 2 * half;
    v2f a, bv;
    a.x  = Vs[li][y0];
    a.y  = Vs[li][y0 + 1];
    bv.x = P[wid][li][y0];
    bv.y = P[wid][li][y0 + 1];
    acc = wmma4(a, bv, acc);
  }
  for (int r = 0; r < 8; ++r) {
    int c = r + 8 * half;
    int f = f0 + li;
    fof[(((size_t)b * DC_ + c) * F_ + f) * TY_ + 64 + t] = acc[r];
  }
}

// ---------------------------------------------------------------------------
// K4a: softmax-over-f statistics via WMMA + deterministic ONLINE softmax.
// Wave owns a 16t x 16y tile; loops f=0..255 sequentially (fixed order, no
// atomics), 4 WMMA per f, running (m,z) with rescaling. Writes M,Z (b,t,y).
// Grid: b(4) * t-tiles(16) * y-pairs(10); block = 64 threads (2 waves).
// ---------------------------------------------------------------------------
__global__ void k_tstats(const float* __restrict__ qt,
                         const float* __restrict__ ktf,
                         float* __restrict__ Mbuf, float* __restrict__ Zbuf) {
  const int wid = threadIdx.x >> 5, lane = threadIdx.x & 31;
  const int li = lane & 15, half = lane >> 4;
  int yp = blockIdx.x % 10;
  int tt = (blockIdx.x / 10) & 15;
  int b  = blockIdx.x / 160;
  int yt = yp * 2 + wid;                  // 0..19

  float m[8], z[8];
  for (int r = 0; r < 8; ++r) { m[r] = -3.4e38f; z[r] = 0.f; }

  for (int f = 0; f < F_; ++f) {
    v8f acc = {};
    for (int kw = 0; kw < 4; ++kw) {
      int c0 = 4 * kw + 2 * half;
      v2f a, bv;
      a.x  = qt [(((size_t)b * DC_ + c0)     * F_ + f) * T_  + tt * 16 + li];
      a.y  = qt [(((size_t)b * DC_ + c0 + 1) * F_ + f) * T_  + tt * 16 + li];
      bv.x = ktf[(((size_t)b * DC_ + c0)     * F_ + f) * TY_ + yt * 16 + li];
      bv.y = ktf[(((size_t)b * DC_ + c0 + 1) * F_ + f) * TY_ + yt * 16 + li];
      acc = wmma4(a, bv, acc);
    }
    for (int r = 0; r < 8; ++r) {
      float s  = acc[r] * 0.25f;
      float mn = fmaxf(m[r], s);
      z[r] = z[r] * __expf(m[r] - mn) + __expf(s - mn);
      m[r] = mn;
    }
  }
  for (int r = 0; r < 8; ++r) {
    int t = tt * 16 + r + 8 * half;
    int y = yt * 16 + li;
    size_t o = ((size_t)b * T_ + t) * TY_ + y;
    Mbuf[o] = m[r];
    Zbuf[o] = z[r];
  }
}

// ---------------------------------------------------------------------------
// K4b: time attention output. Block = 64 threads (2 waves), (b, f, 32 t's).
// Async-DMA the 16x320 V2 tile (fof rows, contiguous in y) into LDS at entry.
// Recompute score tiles with WMMA, P = exp(s - M)/Z -> LDS (32x320 = 40KB),
// then out[c,t] = sum_y V2[c,y] P[t,y]  (K=320 -> 80 WMMA per t-tile).
// ---------------------------------------------------------------------------
__global__ void k_tattn(const float* __restrict__ qt,
                        const float* __restrict__ ktf,
                        const float* __restrict__ fof,
                        const float* __restrict__ Mbuf,
                        const float* __restrict__ Zbuf,
                        float* __restrict__ tout) {
  __shared__ float P[32][320];     // 40 KB
  __shared__ float Vs[16][320];    // 20 KB
  const int wid = threadIdx.x >> 5, lane = threadIdx.x & 31;
  const int li = lane & 15, half = lane >> 4;
  int tch = blockIdx.x & 7;
  int f   = (blockIdx.x >> 3) & 255;
  int b   = blockIdx.x >> 11;
  int tbase = tch * 32;

  // kick off async V2-tile DMA: 16 rows x 320 floats (1280 B/row, contiguous)
  {
    const float* src = fof + ((size_t)b * DC_ * F_ + f) * TY_;
    for (int i = 0; i < 20; ++i) {            // 1280 b128 / 64 threads
      int e4 = threadIdx.x + 64 * i;
      int c = e4 / 80, y4 = e4 % 80;
      async_copy_b128(&Vs[c][y4 * 4], src + (size_t)c * F_ * TY_ + y4 * 4);
    }
  }

  // phase 1: P tiles (2 t-tiles x 20 y-tiles); Q operand hoisted per t-tile
  for (int tt = 0; tt < 2; ++tt) {
    v2f qa[4];
    for (int kw = 0; kw < 4; ++kw) {
      int c0 = 4 * kw + 2 * half;
      qa[kw].x = qt[(((size_t)b * DC_ + c0)     * F_ + f) * T_ + tbase + tt * 16 + li];
      qa[kw].y = qt[(((size_t)b * DC_ + c0 + 1) * F_ + f) * T_ + tbase + tt * 16 + li];
    }
    for (int yt = wid; yt < 20; yt += 2) {
      v8f acc = {};
      for (int kw = 0; kw < 4; ++kw) {
        int c0 = 4 * kw + 2 * half;
        v2f bv;
        bv.x = ktf[(((size_t)b * DC_ + c0)     * F_ + f) * TY_ + yt * 16 + li];
        bv.y = ktf[(((size_t)b * DC_ + c0 + 1) * F_ + f) * TY_ + yt * 16 + li];
        acc = wmma4(qa[kw], bv, acc);
      }
      for (int r = 0; r < 8; ++r) {
        int tl = tt * 16 + r + 8 * half;     // local t 0..31
        int y  = yt * 16 + li;
        size_t o = ((size_t)b * T_ + tbase + tl) * TY_ + y;
        P[tl][y] = __expf(acc[r] * 0.25f - Mbuf[o]) / Zbuf[o];
      }
    }
  }
  async_wait0();        // V2 tile resident
  __syncthreads();

  // phase 2: wave wid -> its own t-tile; out[c,t] over K=320, A from LDS
  v8f acc = {};
  for (int kw = 0; kw < 80; ++kw) {
    int y0 = 4 * kw + 2 * half;
    v2f a, bv;
    a.x  = Vs[li][y0];
    a.y  = Vs[li][y0 + 1];
    bv.x = P[wid * 16 + li][y0];
    bv.y = P[wid * 16 + li][y0 + 1];
    acc = wmma4(a, bv, acc);
  }
  for (int r = 0; r < 8; ++r) {
    int c = r + 8 * half;
    int t = tbase + wid * 16 + li;
    tout[(((size_t)b * DC_ + c) * F_ + f) * T_ + t] = acc[r];
  }
}

// ---------------------------------------------------------------------------
// K5: proj conv (M=64, K=16) + BN + PReLU + residual -> d_out[0:B*C*F*T)
// ---------------------------------------------------------------------------
__global__ void k_proj(const float* __restrict__ tout,
                       const float* __restrict__ pw,       // (64,16)
                       const float* __restrict__ scaleP,
                       const float* __restrict__ shiftP,
                       const float* __restrict__ alphaP,
                       const float* __restrict__ inp,
                       float* __restrict__ out) {
  const int wid = threadIdx.x >> 5, lane = threadIdx.x & 31;
  const int li = lane & 15, half = lane >> 4;
  int gid = blockIdx.x * 8 + wid;       // 65536 wave tiles
  int mt = gid & 3;
  int nt = gid >> 2;
  int n0 = nt * 16;
  int t0 = n0 & 255, f = (n0 >> 8) & 255, b = n0 >> 16;

  v8f acc = {};
  for (int kw = 0; kw < 4; ++kw) {
    int c0 = 4 * kw + 2 * half;
    v2f a, bv;
    a.x  = pw[(mt * 16 + li) * DC_ + c0];
    a.y  = pw[(mt * 16 + li) * DC_ + c0 + 1];
    bv.x = tout[(((size_t)b * DC_ + c0)     * F_ + f) * T_ + t0 + li];
    bv.y = tout[(((size_t)b * DC_ + c0 + 1) * F_ + f) * T_ + t0 + li];
    acc = wmma4(a, bv, acc);
  }
  for (int r = 0; r < 8; ++r) {
    int o = mt * 16 + r + 8 * half;
    float y = acc[r] * scaleP[o] + shiftP[o];
    y = (y >= 0.f) ? y : alphaP[o] * y;
    size_t idx = (((size_t)b * C_ + o) * F_ + f) * T_ + t0 + li;
    out[idx] = y + inp[idx];
  }
}

// ---------------------------------------------------------------------------
// K6: new_cache = concat(kt[..,1:], f_out[..,1:]) -> d_out tail (B,32,F,319)
// ---------------------------------------------------------------------------
__global__ void k_cacheout(const float* __restrict__ ktf,
                           const float* __restrict__ fof,
                           float* __restrict__ oc) {
  size_t i = (size_t)blockIdx.x * 256 + threadIdx.x;
  const size_t N = (size_t)B_ * 32 * F_ * 319;
  if (i >= N) return;
  int j = (int)(i % 319); size_t r = i / 319;
  int f = (int)(r & 255); r >>= 8;
  int ch = (int)(r & 31); int b = (int)(r >> 5);
  const float* src = (ch < 16) ? ktf : fof;
  int cc = ch & 15;
  oc[i] = src[(((size_t)b * DC_ + cc) * F_ + f) * TY_ + 1 + j];
}

// ---------------------------------------------------------------------------
extern "C" void kernel_launch(void* const* d_in, const int* in_sizes, int n_in,
                              void* d_out, int out_size, void* d_ws, size_t ws_size,
                              hipStream_t stream) {
  (void)in_sizes; (void)n_in; (void)out_size; (void)ws_size;
  const float* inp    = (const float*)d_in[0];
  const float* cache  = (const float*)d_in[1];
  const float* fqkv_w = (const float*)d_in[2];
  const float* fqkv_g = (const float*)d_in[3];
  const float* fqkv_b = (const float*)d_in[4];
  const float* fqkv_m = (const float*)d_in[5];
  const float* fqkv_v = (const float*)d_in[6];
  const float* fqkv_a = (const float*)d_in[7];
  const float* tqk_w  = (const float*)d_in[8];
  const float* tqk_g  = (const float*)d_in[9];
  const float* tqk_b  = (const float*)d_in[10];
  const float* tqk_m  = (const float*)d_in[11];
  const float* tqk_v  = (const float*)d_in[12];
  const float* tqk_a  = (const float*)d_in[13];
  const float* proj_w = (const float*)d_in[14];
  const float* proj_g = (const float*)d_in[15];
  const float* proj_b = (const float*)d_in[16];
  const float* proj_m = (const float*)d_in[17];
  const float* proj_v = (const float*)d_in[18];
  const float* proj_a = (const float*)d_in[19];

  float* ws = (float*)d_ws;
  size_t o = 0;
  float* qfT  = ws + o; o += (size_t)B_ * DC_ * F_ * T_;   // (b,t,c,f)
  float* kfT  = ws + o; o += (size_t)B_ * DC_ * F_ * T_;   // (b,t,c,f)
  float* vvT  = ws + o; o += (size_t)B_ * DC_ * F_ * T_;   // (b,t,c,f)
  float* qt   = ws + o; o += (size_t)B_ * DC_ * F_ * T_;   // (b,c,f,t)
  float* ktf  = ws + o; o += (size_t)B_ * DC_ * F_ * TY_;  // 5,242,880
  float* fof  = ws + o; o += (size_t)B_ * DC_ * F_ * TY_;
  float* tout = ws + o; o += (size_t)B_ * DC_ * F_ * T_;
  float* Mbuf = ws + o; o += (size_t)B_ * T_ * TY_;        // 327,680
  float* Zbuf = ws + o; o += (size_t)B_ * T_ * TY_;
  float* scaleA = ws + o; o += 80;
  float* shiftA = ws + o; o += 80;
  float* alphaA = ws + o; o += 80;
  float* scaleP = ws + o; o += 64;
  float* shiftP = ws + o; o += 64;
  float* alphaP = ws + o; o += 64;

  k_params<<<1, 256, 0, stream>>>(fqkv_g, fqkv_b, fqkv_m, fqkv_v, fqkv_a,
                                  tqk_g, tqk_b, tqk_m, tqk_v, tqk_a,
                                  proj_g, proj_b, proj_m, proj_v, proj_a,
                                  scaleA, shiftA, alphaA, scaleP, shiftP, alphaP);

  // 5 M-tiles * 16384 N-tiles = 81920 wave tiles / 8 waves per block
  k_qkv<<<10240, 256, 0, stream>>>(inp, fqkv_w, tqk_w, scaleA, shiftA, alphaA,
                                   qfT, kfT, vvT, qt, ktf);

  {
    size_t n = (size_t)B_ * 32 * F_ * 64;
    k_cachein<<<(unsigned)((n + 255) / 256), 256, 0, stream>>>(cache, ktf, fof);
  }

  // b(4) * t(256) * fblocks(8)
  k_fattn<<<8192, 64, 0, stream>>>(qfT, kfT, vvT, fof);

  // b(4) * t-tiles(16) * y-pairs(10)
  k_tstats<<<640, 64, 0, stream>>>(qt, ktf, Mbuf, Zbuf);

  // b(4) * f(256) * tchunks(8)
  k_tattn<<<8192, 64, 0, stream>>>(qt, ktf, fof, Mbuf, Zbuf, tout);

  // 4 M-tiles * 16384 N-tiles = 65536 wave tiles / 8 per block
  k_proj<<<8192, 256, 0, stream>>>(tout, proj_w, scaleP, shiftP, alphaP, inp,
                                   (float*)d_out);

  {
    size_t n = (size_t)B_ * 32 * F_ * 319;
    float* oc = (float*)d_out + (size_t)B_ * C_ * F_ * T_;
    k_cacheout<<<(unsigned)((n + 255) / 256), 256, 0, stream>>>(ktf, fof, oc);
  }
}